// RhymeLoss_80152679678796
// MI455X (gfx1250) — compile-verified
//
#include <hip/hip_runtime.h>
#include <math.h>

typedef __attribute__((ext_vector_type(2))) float v2f;
typedef __attribute__((ext_vector_type(8))) float v8f;

#define BATCHES      4096
#define NLINES       32
#define DIM          256
#define WAVES_PER_BLK 8
#define EPS_F        1e-8f

// One wave32 per batch: compute G = A * A^T for the 32x256 block via
// v_wmma_f32_16x16x4_f32 (exact f32), recover norms from the diagonal,
// apply the contrastive rhyme loss over the strict upper triangle,
// wave-reduce, write one partial per batch.
__global__ __launch_bounds__(WAVES_PER_BLK * 32)
void rhyme_gram_kernel(const float* __restrict__ emb,
                       const int* __restrict__ scheme,
                       float* __restrict__ partial) {
    const int lane = threadIdx.x & 31;
    const int wave = threadIdx.x >> 5;
    const int batch = blockIdx.x * WAVES_PER_BLK + wave;

    __shared__ float diagLds[WAVES_PER_BLK][NLINES];
    __shared__ int   schLds[WAVES_PER_BLK][NLINES];

    const float* A = emb + (size_t)batch * (NLINES * DIM);

    // f32 WMMA A/B fragment addressing: lane L<16 -> (row=L, k+0..1),
    // lane L>=16 -> (row=L-16, k+2..3). Gram trick: same regs feed SRC0 & SRC1.
    const int row  = lane & 15;
    const int koff = (lane >> 4) << 1;  // 0 or 2

    const float* p0 = A + row * DIM + koff;          // rows 0..15 fragment
    const float* p1 = A + (row + 16) * DIM + koff;   // rows 16..31 fragment

    v8f c00 = {}; // G[0:16, 0:16]
    v8f c01 = {}; // G[0:16, 16:32]
    v8f c11 = {}; // G[16:32, 16:32]

    #pragma unroll 8
    for (int k = 0; k < DIM; k += 4) {
        v2f a0 = *(const v2f*)(p0 + k);
        v2f a1 = *(const v2f*)(p1 + k);
        c00 = __builtin_amdgcn_wmma_f32_16x16x4_f32(false, a0, false, a0,
                                                    (short)0, c00, false, false);
        c01 = __builtin_amdgcn_wmma_f32_16x16x4_f32(false, a0, false, a1,
                                                    (short)0, c01, false, false);
        c11 = __builtin_amdgcn_wmma_f32_16x16x4_f32(false, a1, false, a1,
                                                    (short)0, c11, false, false);
    }

    // Stash rhyme scheme (32 ints) and gram diagonals (32 floats) in LDS.
    schLds[wave][lane] = scheme[(size_t)batch * NLINES + lane];

    const int mbase = (lane >> 4) << 3;  // 0 or 8
    const int ncol  = lane & 15;
    #pragma unroll
    for (int v = 0; v < 8; ++v) {
        const int m = mbase + v;             // tile-local row 0..15
        if (m == ncol) {                     // this lane holds (m,m)
            diagLds[wave][m]      = c00[v];  // global rows 0..15
            diagLds[wave][16 + m] = c11[v];  // global rows 16..31
        }
    }
    __syncthreads();

    // Column-side norms / schemes (fixed per lane).
    const float invN0 = 1.0f / fmaxf(sqrtf(diagLds[wave][ncol]),      EPS_F);
    const float invN1 = 1.0f / fmaxf(sqrtf(diagLds[wave][16 + ncol]), EPS_F);
    const int   sN0   = schLds[wave][ncol];
    const int   sN1   = schLds[wave][16 + ncol];

    float sum = 0.0f;
    #pragma unroll
    for (int v = 0; v < 8; ++v) {
        const int m = mbase + v;
        const float invM0 = 1.0f / fmaxf(sqrtf(diagLds[wave][m]),      EPS_F);
        const float invM1 = 1.0f / fmaxf(sqrtf(diagLds[wave][16 + m]), EPS_F);
        const int   sM0   = schLds[wave][m];
        const int   sM1   = schLds[wave][16 + m];

        // Tile D00: pair (m, ncol), strict upper only.
        if (m < ncol) {
            const float sim = c00[v] * invM0 * invN0;
            sum += (sM0 == sN0) ? (1.0f - sim) : fmaxf(sim - 0.5f, 0.0f);
        }
        // Tile D01: pair (m, 16+ncol) -> always i < j.
        {
            const float sim = c01[v] * invM0 * invN1;
            sum += (sM0 == sN1) ? (1.0f - sim) : fmaxf(sim - 0.5f, 0.0f);
        }
        // Tile D11: pair (16+m, 16+ncol), strict upper only.
        if (m < ncol) {
            const float sim = c11[v] * invM1 * invN1;
            sum += (sM1 == sN1) ? (1.0f - sim) : fmaxf(sim - 0.5f, 0.0f);
        }
    }

    // Wave32 butterfly reduction.
    #pragma unroll
    for (int off = 16; off > 0; off >>= 1)
        sum += __shfl_xor(sum, off, 32);

    if (lane == 0)
        partial[batch] = sum;
}

// Deterministic final reduction: fixed-order strided accumulate + tree.
__global__ __launch_bounds__(256)
void rhyme_reduce_kernel(const float* __restrict__ partial,
                         float* __restrict__ out) {
    __shared__ float s[256];
    float acc = 0.0f;
    for (int i = threadIdx.x; i < BATCHES; i += 256)
        acc += partial[i];
    s[threadIdx.x] = acc;
    __syncthreads();
    #pragma unroll
    for (int stride = 128; stride > 0; stride >>= 1) {
        if (threadIdx.x < stride)
            s[threadIdx.x] += s[threadIdx.x + stride];
        __syncthreads();
    }
    if (threadIdx.x == 0) {
        const float inv_count =
            1.0f / ((float)BATCHES * (float)(NLINES * (NLINES - 1) / 2));
        out[0] = s[0] * inv_count;
    }
}

extern "C" void kernel_launch(void* const* d_in, const int* in_sizes, int n_in,
                              void* d_out, int out_size, void* d_ws, size_t ws_size,
                              hipStream_t stream) {
    const float* emb    = (const float*)d_in[0];   // [4096, 32, 256] f32
    const int*   scheme = (const int*)d_in[1];     // [4096, 32] i32
    float* out     = (float*)d_out;                // scalar f32
    float* partial = (float*)d_ws;                 // 4096 floats of scratch

    const int blocks = BATCHES / WAVES_PER_BLK;    // 512
    rhyme_gram_kernel<<<blocks, WAVES_PER_BLK * 32, 0, stream>>>(emb, scheme, partial);
    rhyme_reduce_kernel<<<1, 256, 0, stream>>>(partial, out);
}